// _CTE_60318520705439
// MI455X (gfx1250) — compile-verified
//
#include <hip/hip_runtime.h>
#include <hip/hip_bf16.h>
#include <stdint.h>

#ifndef __has_builtin
#define __has_builtin(x) 0
#endif

#define HD 4096
#define YDIM 1024
#define DLT 0.1
#define NSTEPS 8
#define TR_SLICES 32
#define Y_KSPLIT 16

typedef float v2f __attribute__((ext_vector_type(2)));
typedef float v8f __attribute__((ext_vector_type(8)));
typedef int   v4i __attribute__((ext_vector_type(4)));

typedef __attribute__((address_space(1))) v4i* gas_v4i_ptr;
typedef __attribute__((address_space(3))) v4i* las_v4i_ptr;

// ---- CDNA5 async global->LDS copy (16B per lane), guarded ----
__device__ __forceinline__ void async_copy16(void* lds_dst, const void* g_src) {
#if __has_builtin(__builtin_amdgcn_global_load_async_to_lds_b128)
  __builtin_amdgcn_global_load_async_to_lds_b128(
      (gas_v4i_ptr)g_src, (las_v4i_ptr)lds_dst, 0, 0);
#else
  *(float4*)lds_dst = *(const float4*)g_src;
#endif
}
__device__ __forceinline__ void async_wait_all() {
#if __has_builtin(__builtin_amdgcn_s_wait_asynccnt)
  __builtin_amdgcn_s_wait_asynccnt(0);
#endif
}

// pq[2i]=h[i]=p0 ; acc[2i]=h[i] (c0 = 1)
__global__ __launch_bounds__(256) void init_h_kernel(const float* __restrict__ h,
                                                     float* __restrict__ pq,
                                                     float* __restrict__ acc) {
  int i = blockIdx.x * 256 + threadIdx.x;
  float v = h[i];
  pq[2 * i] = v;
  acc[2 * i] = v;
}

// b[r] = sum_j Bw[r][j]*udu[j], udu=[du(1024), u(3072)]; pq[2r+1]=b, acc[2r+1]=DLT*b
__global__ __launch_bounds__(256) void bgemv_kernel(const float* __restrict__ Bw,
                                                    const float* __restrict__ u,
                                                    const float* __restrict__ du,
                                                    float* __restrict__ pq,
                                                    float* __restrict__ acc) {
  const int wave = threadIdx.x >> 5;
  const int lane = threadIdx.x & 31;
#pragma unroll
  for (int rr = 0; rr < 2; ++rr) {
    const int r = blockIdx.x * 16 + wave + rr * 8;
    const float* Br = Bw + (size_t)r * HD;
    float s = 0.f;
    for (int it = 0; it < 8; ++it) {  // cols 0..1023 -> du
      const int c = it * 128 + lane * 4;
      float4 w = *(const float4*)(Br + c);
      float4 v = *(const float4*)(du + c);
      s += w.x * v.x + w.y * v.y + w.z * v.z + w.w * v.w;
    }
    for (int it = 8; it < 32; ++it) {  // cols 1024..4095 -> u
      const int c = it * 128 + lane * 4;
      float4 w = *(const float4*)(Br + c);
      float4 v = *(const float4*)(u + (c - 1024));
      s += w.x * v.x + w.y * v.y + w.z * v.z + w.w * v.w;
    }
    for (int m = 16; m >= 1; m >>= 1) s += __shfl_xor(s, m, 32);
    if (lane == 0) {
      pq[2 * r + 1] = s;
      acc[2 * r + 1] = (float)DLT * s;  // d0 = DELTA
    }
  }
}

// fwd[2r+{0,1}] = sum_c W[r][c]*pq[2c+{0,1}]  (row dots; wave-per-row, DPP reduce)
__global__ __launch_bounds__(256) void fwd_gemv(const float* __restrict__ W,
                                                const float* __restrict__ pq,
                                                float* __restrict__ fwd) {
  __shared__ __align__(16) float spq[2 * HD];  // 32KB staged copy of pq
#pragma unroll
  for (int j = 0; j < 8; ++j) {
    int off = (j * 256 + threadIdx.x) * 4;
    async_copy16(&spq[off], pq + off);
  }
  async_wait_all();
  __syncthreads();

  const int wave = threadIdx.x >> 5;
  const int lane = threadIdx.x & 31;
#pragma unroll
  for (int rr = 0; rr < 2; ++rr) {
    const int r = blockIdx.x * 16 + wave + rr * 8;
    const float* Wr = W + (size_t)r * HD;
    float ap = 0.f, aq = 0.f;
#pragma unroll 4
    for (int it = 0; it < 32; ++it) {
      const int c = it * 128 + lane * 4;
      float4 w = *(const float4*)(Wr + c);
      float4 v0 = *(const float4*)(&spq[2 * c]);      // p,q,p,q
      float4 v1 = *(const float4*)(&spq[2 * c + 4]);
      ap += w.x * v0.x + w.y * v0.z + w.z * v1.x + w.w * v1.z;
      aq += w.x * v0.y + w.y * v0.w + w.z * v1.y + w.w * v1.w;
    }
    for (int m = 16; m >= 1; m >>= 1) {
      ap += __shfl_xor(ap, m, 32);
      aq += __shfl_xor(aq, m, 32);
    }
    if (lane == 0) {
      fwd[2 * r] = ap;
      fwd[2 * r + 1] = aq;
    }
  }
}

// tr_slice[rb][2c+{0,1}] = sum_{r in rowblock} W[r][c]*pq[2r+{0,1}] (column-owned, no reductions)
__global__ __launch_bounds__(256) void tr_gemv(const float* __restrict__ W,
                                               const float* __restrict__ pq,
                                               float* __restrict__ tr) {
  __shared__ __align__(16) float spq[256];  // pq pairs of this row block
  const int cb = blockIdx.x;                // 0..3  (1024 cols each)
  const int rb = blockIdx.y;                // 0..31 (128 rows each)
  const int r0 = rb * 128;
  const int c = cb * 1024 + threadIdx.x * 4;
  if (threadIdx.x < 64) async_copy16(&spq[threadIdx.x * 4], pq + 2 * r0 + threadIdx.x * 4);
  async_wait_all();
  __syncthreads();

  float a0p = 0, a0q = 0, a1p = 0, a1q = 0, a2p = 0, a2q = 0, a3p = 0, a3q = 0;
#pragma unroll 4
  for (int rr = 0; rr < 128; ++rr) {
    float4 w = *(const float4*)(W + (size_t)(r0 + rr) * HD + c);
    float pr = spq[2 * rr], qr = spq[2 * rr + 1];
    a0p += w.x * pr; a0q += w.x * qr;
    a1p += w.y * pr; a1q += w.y * qr;
    a2p += w.z * pr; a2q += w.z * qr;
    a3p += w.w * pr; a3q += w.w * qr;
  }
  float* out = tr + (size_t)rb * (2 * HD) + 2 * c;
  float4 o0 = {a0p, a0q, a1p, a1q};
  float4 o1 = {a2p, a2q, a3p, a3q};
  *(float4*)(out) = o0;
  *(float4*)(out + 4) = o1;
}

// p_k = 0.5*(Wv - W^T v); acc += (ck,dk)*p_k; pq = p_k
__global__ __launch_bounds__(256) void combine_kernel(float* __restrict__ pq,
                                                      float* __restrict__ acc,
                                                      const float* __restrict__ fwd,
                                                      const float* __restrict__ tr,
                                                      float ck, float dk) {
  int i = blockIdx.x * 256 + threadIdx.x;  // 0..4095
  float2 f = *(const float2*)(fwd + 2 * i);
  float tp = 0.f, tq = 0.f;
#pragma unroll 8
  for (int s = 0; s < TR_SLICES; ++s) {
    float2 t = *(const float2*)(tr + (size_t)s * (2 * HD) + 2 * i);
    tp += t.x;
    tq += t.y;
  }
  float np = 0.5f * (f.x - tp);
  float nq = 0.5f * (f.y - tq);
  acc[2 * i] += ck * np;
  acc[2 * i + 1] += dk * nq;
  pq[2 * i] = np;
  pq[2 * i + 1] = nq;
}

// hnew[i] = rec[i] + inp[i]
__global__ __launch_bounds__(256) void hnew_kernel(const float* __restrict__ acc,
                                                   float* __restrict__ hnew) {
  int i = blockIdx.x * 256 + threadIdx.x;  // 0..4095
  float2 a = *(const float2*)(acc + 2 * i);
  hnew[i] = a.x + a.y;
}

// y partials via v_wmma_f32_16x16x4_f32:
//   A[m][k] = Cw[r0+m][k]*hnew[k] (A 16x4 layout documented: lane m=l&15, K=(l>>4)*2+{0,1})
//   B = all-ones (value identical in every slot -> immune to B fragment layout details)
//   => D[m][n] = sum_k Cw[r0+m][k]*hnew[k] for every n; extract per documented D layout.
// Software-pipelined: 8 independent (cw,hn) pair loads per group, then 8 chained WMMAs.
__global__ __launch_bounds__(256) void ywmma_kernel(const float* __restrict__ Cw,
                                                    const float* __restrict__ hnew,
                                                    float* __restrict__ ypart) {
  const int wave = threadIdx.x >> 5;
  const int lane = threadIdx.x & 31;
  const int unit = blockIdx.x * 8 + wave;  // 0..1023
  const int rowtile = unit >> 4;           // 0..63
  const int ks = unit & 15;                // 0..15
  const int r0 = rowtile * 16;
  const int row = lane & 15;
  const int khalf = (lane >> 4) << 1;  // 0 or 2
  const float* crow = Cw + (size_t)(r0 + row) * HD;

  v8f d = {0.f, 0.f, 0.f, 0.f, 0.f, 0.f, 0.f, 0.f};
  const v2f bones = {1.0f, 1.0f};
  const int kbeg = ks * (HD / Y_KSPLIT);  // 256 k-values per unit
  for (int kk = 0; kk < 8; ++kk) {        // 8 groups of 8 WMMAs (32 k per group)
    const int kbase = kbeg + kk * 32 + khalf;
    float2 cw[8], hn[8];
#pragma unroll
    for (int j = 0; j < 8; ++j) {
      cw[j] = *(const float2*)(crow + kbase + j * 4);
      hn[j] = *(const float2*)(hnew + kbase + j * 4);
    }
#pragma unroll
    for (int j = 0; j < 8; ++j) {
      v2f a;
      a.x = cw[j].x * hn[j].x;  // C[i,k] * h_new[k]
      a.y = cw[j].y * hn[j].y;
      d = __builtin_amdgcn_wmma_f32_16x16x4_f32(false, a, false, bones, (short)0, d,
                                                false, false);
    }
  }
  float* out = ypart + (size_t)ks * YDIM;
  if (lane == 0) {
#pragma unroll
    for (int v = 0; v < 8; ++v) out[r0 + v] = d[v];  // rows m=0..7
  } else if (lane == 16) {
#pragma unroll
    for (int v = 0; v < 8; ++v) out[r0 + 8 + v] = d[v];  // rows m=8..15
  }
}

__global__ __launch_bounds__(256) void ycombine_kernel(const float* __restrict__ ypart,
                                                       float* __restrict__ y) {
  int i = blockIdx.x * 256 + threadIdx.x;  // 0..1023
  float s = 0.f;
#pragma unroll
  for (int ks = 0; ks < Y_KSPLIT; ++ks) s += ypart[(size_t)ks * YDIM + i];
  y[i] = s;
}

extern "C" void kernel_launch(void* const* d_in, const int* in_sizes, int n_in,
                              void* d_out, int out_size, void* d_ws, size_t ws_size,
                              hipStream_t stream) {
  (void)in_sizes; (void)n_in; (void)out_size; (void)ws_size;
  const float* u  = (const float*)d_in[0];  // [3072]
  const float* du = (const float*)d_in[1];  // [1024]
  const float* h  = (const float*)d_in[2];  // [4096]
  const float* W  = (const float*)d_in[3];  // [4096,4096]
  const float* Bw = (const float*)d_in[4];  // [4096,4096]
  const float* Cw = (const float*)d_in[5];  // [1024,4096]
  float* y = (float*)d_out;                 // [1024]

  float* ws = (float*)d_ws;
  float* pq    = ws;                         // [2*H] interleaved (p,q)
  float* acc   = ws + 2 * HD;                // [2*H] interleaved (rec, inp)
  float* fwd   = ws + 4 * HD;                // [2*H]
  float* tr    = ws + 6 * HD;                // [TR_SLICES][2*H]
  float* ypart = tr + (size_t)TR_SLICES * 2 * HD;   // [Y_KSPLIT][YDIM]
  float* hnew  = ypart + (size_t)Y_KSPLIT * YDIM;   // [H]

  init_h_kernel<<<HD / 256, 256, 0, stream>>>(h, pq, acc);
  bgemv_kernel<<<HD / 16, 256, 0, stream>>>(Bw, u, du, pq, acc);

  double c = 1.0, dcoef = DLT;  // c_k = D^k/k!, d_k = D^(k+1)/(k+1)!
  for (int k = 1; k <= NSTEPS; ++k) {
    c *= DLT / (double)k;
    dcoef *= DLT / (double)(k + 1);
    fwd_gemv<<<HD / 16, 256, 0, stream>>>(W, pq, fwd);
    tr_gemv<<<dim3(4, TR_SLICES), 256, 0, stream>>>(W, pq, tr);
    combine_kernel<<<HD / 256, 256, 0, stream>>>(pq, acc, fwd, tr, (float)c, (float)dcoef);
  }

  hnew_kernel<<<HD / 256, 256, 0, stream>>>(acc, hnew);
  ywmma_kernel<<<(YDIM / 16) * Y_KSPLIT / 8, 256, 0, stream>>>(Cw, hnew, ypart);
  ycombine_kernel<<<YDIM / 256, 256, 0, stream>>>(ypart, y);
}